// LayerGIN_84602265796937
// MI455X (gfx1250) — compile-verified
//
#include <hip/hip_runtime.h>

typedef __attribute__((ext_vector_type(2))) float v2f;
typedef __attribute__((ext_vector_type(8))) float v8f;

#define NN    4096
#define DIM   256
#define HEADS 2
#define HD    128
#define ADJ_W 128   // u32 words per adjacency row (4096 bits)

// ---------------------------------------------------------------- utilities
__global__ void k_zero_u32(unsigned* __restrict__ p, int n) {
  int i = blockIdx.x * blockDim.x + threadIdx.x;
  if (i < n) p[i] = 0u;
}

__global__ void k_build_adj(const int* __restrict__ ei, int E,
                            unsigned* __restrict__ adj) {
  int e = blockIdx.x * blockDim.x + threadIdx.x;
  if (e >= E) return;
  int r = ei[e];        // edge_index[0][e]
  int c = ei[E + e];    // edge_index[1][e]
  atomicOr(&adj[(size_t)r * ADJ_W + (c >> 5)], 1u << (c & 31));
}

// ------------------------------------------------- fp32 WMMA GEMM (+bias,+eps*resid)
// C[M,N] = A[M,K] @ B[K,N] + bias[N] (+ eps*resid[M,N])
// One wave computes a 32(M) x 64(N) tile: 2 A-fragments x 4 B-fragments,
// so each B fragment feeds two v_wmma ops (10 VMEM loads per 8 WMMA per K-step).
// A fragment (16x4 f32): lanes 0-15 hold K=k..k+1, lanes 16-31 hold K=k+2..k+3.
// B fragment (4x16 f32): v0 = rows k / k+2 striped over lane halves, v1 = rows k+1 / k+3.
__global__ void k_gemm_wmma(const float* __restrict__ A, const float* __restrict__ B,
                            const float* __restrict__ bias,
                            const float* __restrict__ resid,
                            const float* __restrict__ epsp,
                            float* __restrict__ C, int M, int N, int K) {
  const int lane  = threadIdx.x & 31;
  const int half  = lane >> 4;      // 0 or 1
  const int l16   = lane & 15;
  const int tileM = blockIdx.x * 32;
  const int tileN = blockIdx.y * 64;

  const float* Arow0 = A + (size_t)(tileM + l16) * K;
  const float* Arow1 = A + (size_t)(tileM + 16 + l16) * K;

  v8f acc[2][4];
#pragma unroll
  for (int i = 0; i < 2; ++i)
#pragma unroll
    for (int j = 0; j < 4; ++j)
      acc[i][j] = (v8f){0.f,0.f,0.f,0.f,0.f,0.f,0.f,0.f};

  for (int k = 0; k < K; k += 4) {
    float2 a02 = *(const float2*)(Arow0 + k + 2 * half);
    float2 a12 = *(const float2*)(Arow1 + k + 2 * half);
    v2f a0; a0[0] = a02.x; a0[1] = a02.y;
    v2f a1; a1[0] = a12.x; a1[1] = a12.y;
    const float* Bk = B + (size_t)(k + 2 * half) * N + tileN + l16;
#pragma unroll
    for (int j = 0; j < 4; ++j) {
      v2f b; b[0] = Bk[j * 16]; b[1] = Bk[N + j * 16];
      acc[0][j] = __builtin_amdgcn_wmma_f32_16x16x4_f32(
          false, a0, false, b, (short)0, acc[0][j], false, false);
      acc[1][j] = __builtin_amdgcn_wmma_f32_16x16x4_f32(
          false, a1, false, b, (short)0, acc[1][j], false, false);
    }
  }

  float epsv = 0.f;
  if (resid) epsv = epsp[0];

#pragma unroll
  for (int i = 0; i < 2; ++i) {
#pragma unroll
    for (int j = 0; j < 4; ++j) {
#pragma unroll
      for (int r = 0; r < 8; ++r) {
        int row = tileM + i * 16 + r + 8 * half;   // C layout: vgpr r -> M=r / M=8+r
        int col = tileN + j * 16 + l16;            // lane -> N
        float val = acc[i][j][r] + bias[col];
        if (resid) val += epsv * resid[(size_t)row * N + col];
        C[(size_t)row * N + col] = val;
      }
    }
  }
}

// ------------------------------------------------- sparse masked attention
// One workgroup (64 thr = 2 waves) per node; wave h handles head h.
// Lane owns 4 contiguous dims of hd=128. Online (flash) softmax over set bits
// of the node's adjacency row (bitset naturally dedupes duplicate edges,
// matching the reference mask semantics).
__global__ void k_attn(const float* __restrict__ Q, const float* __restrict__ Kb,
                       const float* __restrict__ Vb, const unsigned* __restrict__ adj,
                       float* __restrict__ out) {
  __shared__ unsigned rowbits[ADJ_W];
  const int n    = blockIdx.x;
  const int h    = threadIdx.x >> 5;
  const int lane = threadIdx.x & 31;

  for (int i = threadIdx.x; i < ADJ_W; i += 64)
    rowbits[i] = adj[(size_t)n * ADJ_W + i];
  __syncthreads();

  const int base = h * HD + lane * 4;
  const float4 q4 = *(const float4*)(Q + (size_t)n * DIM + base);

  const float scale = 0.088388347648318447f;  // 1/sqrt(128)
  float m = -3.0e38f, l = 0.f;
  float4 acc = make_float4(0.f, 0.f, 0.f, 0.f);

  for (int w = 0; w < ADJ_W; ++w) {
    unsigned bits = rowbits[w];
    while (bits) {
      int b = __ffs(bits) - 1;
      bits &= bits - 1;
      int col = (w << 5) + b;

      const float4 k4 = *(const float4*)(Kb + (size_t)col * DIM + base);
      float p = q4.x * k4.x + q4.y * k4.y + q4.z * k4.z + q4.w * k4.w;
#pragma unroll
      for (int off = 16; off >= 1; off >>= 1) p += __shfl_xor(p, off, 32);
      float s = p * scale;

      const float4 v4 = *(const float4*)(Vb + (size_t)col * DIM + base);
      float mn = fmaxf(m, s);
      float cm = __expf(m - mn);
      float pe = __expf(s - mn);
      l = l * cm + pe;
      acc.x = acc.x * cm + pe * v4.x;
      acc.y = acc.y * cm + pe * v4.y;
      acc.z = acc.z * cm + pe * v4.z;
      acc.w = acc.w * cm + pe * v4.w;
      m = mn;
    }
  }

  float inv = 1.f / l;   // self-loops guarantee l > 0
  float4 o = make_float4(acc.x * inv, acc.y * inv, acc.z * inv, acc.w * inv);
  *(float4*)(out + (size_t)n * DIM + base) = o;
}

// ------------------------------------------------- BatchNorm (deterministic 2-pass)
__global__ void k_bn_partial(const float* __restrict__ Y, float* __restrict__ part) {
  const int t  = threadIdx.x;          // column 0..255
  const int r0 = blockIdx.x * 32;      // 128 blocks x 32 rows
  float s = 0.f, ss = 0.f;
#pragma unroll 4
  for (int r = 0; r < 32; ++r) {
    float v = Y[(size_t)(r0 + r) * DIM + t];
    s += v; ss += v * v;
  }
  part[blockIdx.x * DIM + t]             = s;
  part[128 * DIM + blockIdx.x * DIM + t] = ss;
}

__global__ void k_bn_finalize(const float* __restrict__ part,
                              const float* __restrict__ g, const float* __restrict__ be,
                              float* __restrict__ stats) {
  const int t = threadIdx.x;
  float s = 0.f, ss = 0.f;
  for (int b = 0; b < 128; ++b) {            // fixed order -> deterministic
    s  += part[b * DIM + t];
    ss += part[128 * DIM + b * DIM + t];
  }
  float mean = s * (1.f / NN);
  float var  = ss * (1.f / NN) - mean * mean;   // biased var (jnp.var)
  float sc   = g[t] * rsqrtf(var + 1e-5f);
  stats[t]       = sc;
  stats[DIM + t] = be[t] - mean * sc;
}

__global__ void k_bn_apply_relu(const float* __restrict__ Y,
                                const float* __restrict__ stats,
                                float* __restrict__ Out, int total) {
  int i = blockIdx.x * blockDim.x + threadIdx.x;
  if (i >= total) return;
  int c = i & (DIM - 1);
  float v = Y[i] * stats[c] + stats[DIM + c];
  Out[i] = fmaxf(v, 0.f);
}

// ------------------------------------------------------------------ driver
extern "C" void kernel_launch(void* const* d_in, const int* in_sizes, int n_in,
                              void* d_out, int out_size, void* d_ws, size_t ws_size,
                              hipStream_t stream) {
  const float* x   = (const float*)d_in[0];
  const int*   ei  = (const int*)d_in[1];
  const float* wq  = (const float*)d_in[2];
  const float* bq  = (const float*)d_in[3];
  const float* wk  = (const float*)d_in[4];
  const float* bk  = (const float*)d_in[5];
  const float* wv  = (const float*)d_in[6];
  const float* bv  = (const float*)d_in[7];
  const float* wo  = (const float*)d_in[8];
  const float* bo  = (const float*)d_in[9];
  const float* eps = (const float*)d_in[10];
  const float* w1  = (const float*)d_in[11];
  const float* b1  = (const float*)d_in[12];
  const float* g1  = (const float*)d_in[13];
  const float* be1 = (const float*)d_in[14];
  const float* w2  = (const float*)d_in[15];
  const float* b2  = (const float*)d_in[16];
  const float* g2  = (const float*)d_in[17];
  const float* be2 = (const float*)d_in[18];

  char* ws = (char*)d_ws;
  const size_t MB = 1024u * 1024u;
  float*    Q    = (float*)(ws + 0 * MB);   // reused as out2
  float*    Kb   = (float*)(ws + 4 * MB);   // reused as y1
  float*    Vb   = (float*)(ws + 8 * MB);   // reused as y2
  float*    vagg = (float*)(ws + 12 * MB);
  unsigned* adj  = (unsigned*)(ws + 16 * MB);          // 2 MB bitset
  float*    part = (float*)(ws + 18 * MB);             // 256 KB BN partials
  float*    stats= (float*)(ws + 18 * MB + 256 * 1024);
  float*    out2 = Q;
  float*    y1   = Kb;
  float*    y2   = Vb;

  const int E = in_sizes[1] / 2;
  const int adjN = NN * ADJ_W;

  k_zero_u32<<<(adjN + 255) / 256, 256, 0, stream>>>(adj, adjN);
  k_build_adj<<<(E + 255) / 256, 256, 0, stream>>>(ei, E, adj);

  dim3 gg(NN / 32, DIM / 64);
  k_gemm_wmma<<<gg, 32, 0, stream>>>(x, wq, bq, nullptr, nullptr, Q,  NN, DIM, DIM);
  k_gemm_wmma<<<gg, 32, 0, stream>>>(x, wk, bk, nullptr, nullptr, Kb, NN, DIM, DIM);
  k_gemm_wmma<<<gg, 32, 0, stream>>>(x, wv, bv, nullptr, nullptr, Vb, NN, DIM, DIM);

  k_attn<<<NN, 64, 0, stream>>>(Q, Kb, Vb, adj, vagg);

  k_gemm_wmma<<<gg, 32, 0, stream>>>(vagg, wo, bo, x, eps, out2, NN, DIM, DIM);

  // layer 1: y1 = out2 @ w1 + b1 ; BN ; ReLU (in place)
  k_gemm_wmma<<<gg, 32, 0, stream>>>(out2, w1, b1, nullptr, nullptr, y1, NN, DIM, DIM);
  k_bn_partial<<<128, 256, 0, stream>>>(y1, part);
  k_bn_finalize<<<1, 256, 0, stream>>>(part, g1, be1, stats);
  k_bn_apply_relu<<<(NN * DIM + 255) / 256, 256, 0, stream>>>(y1, stats, y1, NN * DIM);

  // layer 2: y2 = y1 @ w2 + b2 ; BN ; ReLU -> d_out
  k_gemm_wmma<<<gg, 32, 0, stream>>>(y1, w2, b2, nullptr, nullptr, y2, NN, DIM, DIM);
  k_bn_partial<<<128, 256, 0, stream>>>(y2, part);
  k_bn_finalize<<<1, 256, 0, stream>>>(part, g2, be2, stats);
  k_bn_apply_relu<<<(NN * DIM + 255) / 256, 256, 0, stream>>>(y2, stats, (float*)d_out, NN * DIM);
}